// WSDAN_49022756717328
// MI455X (gfx1250) — compile-verified
//
#include <hip/hip_runtime.h>
#include <math.h>

// ---------------- problem constants (from reference) ----------------
#define B_    64
#define C_    768
#define HW_   676          // 26*26
#define M_    32
#define NCLS  200
#define KFC   (M_ * C_)    // 24576
#define EPSN  1e-12f

typedef __attribute__((ext_vector_type(2))) float v2f;
typedef __attribute__((ext_vector_type(8))) float v8f;

// fp32 WMMA: D(16x16) = A(16x4) * B(4x16) + C, wave32.
// 8-arg form: (neg_a, A, neg_b, B, c_mod, C, reuse_a, reuse_b)
__device__ __forceinline__ v8f wmma4(v2f a, v2f b, v8f c) {
  return __builtin_amdgcn_wmma_f32_16x16x4_f32(false, a, false, b, (short)0, c,
                                               false, false);
}

// Operand lane mapping used throughout (consistent with ISA §7.12.2):
//   A: lane L holds A[m0+(L&15)][k0 + 2*(L>>4) + r]   (r = reg 0/1)
//   B: lane L holds B[k0 + 2*(L>>4) + r][n0+(L&15)]
//   D: reg r, lanes 0-15 -> (m0+r, n), lanes 16-31 -> (m0+r+8, n)

// =====================================================================
// Kernel 1: att_m[b] (32x676) = relu( att_w[0:32] (32x768) @ F[b] + bias )
// grid (64, 11), block 256 (8 waves). wave w: m-tile (w&1), hw-tile blk*4+(w>>1)
// =====================================================================
__global__ __launch_bounds__(256) void k_att(const float* __restrict__ F,
                                             const float* __restrict__ W,
                                             const float* __restrict__ bias,
                                             float* __restrict__ att_out) {
  __shared__ float S_w[M_][17];  // att_w k-panel, padded stride
  const int b    = blockIdx.x;
  const int tid  = threadIdx.x;
  const int lane = tid & 31;
  const int w    = tid >> 5;
  const int ln   = lane & 15;
  const int kh   = (lane >> 4) << 1;        // 0 or 2
  const int m0   = (w & 1) * 16;
  const int hwt  = blockIdx.y * 4 + (w >> 1);
  const int hw0  = hwt * 16;
  const int n    = hw0 + ln;
  const bool nok = (hwt < 43) && (n < HW_);
  const float* Fb = F + (size_t)b * C_ * HW_;

  v8f acc = {0.f, 0.f, 0.f, 0.f, 0.f, 0.f, 0.f, 0.f};

  for (int kt = 0; kt < C_ / 16; ++kt) {     // 48 k-tiles
#pragma unroll
    for (int e = 0; e < 2; ++e) {            // stage 32x16 of att_w, coalesced
      int lin = e * 256 + tid;
      int mm = lin >> 4, kk = lin & 15;
      S_w[mm][kk] = W[mm * C_ + kt * 16 + kk];
    }
    __syncthreads();
#pragma unroll
    for (int s = 0; s < 4; ++s) {
      const int kc = s * 4 + kh;             // k within tile for this lane
      v2f a;
      a.x = S_w[m0 + ln][kc];
      a.y = S_w[m0 + ln][kc + 1];
      float b0 = 0.f, b1 = 0.f;
      if (nok) {
        const float* fp = Fb + (size_t)(kt * 16 + kc) * HW_ + n;
        b0 = fp[0];                           // F[k  , n]
        b1 = fp[HW_];                         // F[k+1, n]
      }
      v2f bb; bb.x = b0; bb.y = b1;
      acc = wmma4(a, bb, acc);
    }
    __syncthreads();
  }

  if (hwt < 43 && n < HW_) {
    float* ao = att_out + (size_t)b * M_ * HW_;
#pragma unroll
    for (int r = 0; r < 8; ++r) {
      const int m = m0 + r + ((lane >> 4) << 3);
      float v = acc[r] + bias[m];
      v = v > 0.f ? v : 0.f;                  // relu
      ao[(size_t)m * HW_ + n] = v;
    }
  }
}

// =====================================================================
// Kernel 2: fm_s[b] (32x768) = ssqrt( att_m[b] (32x676) @ F[b]^T / 676 )
// + per-(b,cb) partial sum of squares into ws (exclusive slot, no atomics)
// grid (64, 6 blocks of 128 channels), block 256 (8 waves)
// =====================================================================
__global__ __launch_bounds__(256) void k_fm(const float* __restrict__ F,
                                            const float* __restrict__ att,
                                            float* __restrict__ fm_out,
                                            float* __restrict__ ws) {
  __shared__ float S_f[16][129];   // transposed F tile: S_f[k][c], padded
  __shared__ float S_a[M_][17];    // att k-tile, zero padded past 676
  __shared__ float red[8];
  const int b    = blockIdx.x;
  const int cb   = blockIdx.y;     // channel block of 128
  const int tid  = threadIdx.x;
  const int lane = tid & 31;
  const int w    = tid >> 5;
  const int ln   = lane & 15;
  const int kh   = (lane >> 4) << 1;
  const int m0   = (w & 1) * 16;
  const int ct0  = (w >> 1) * 2;   // two c-tiles per wave
  const float* Fb = F + ((size_t)b * C_ + (size_t)cb * 128) * HW_;
  const float* ab = att + (size_t)b * M_ * HW_;

  v8f acc0 = {0.f, 0.f, 0.f, 0.f, 0.f, 0.f, 0.f, 0.f};
  v8f acc1 = {0.f, 0.f, 0.f, 0.f, 0.f, 0.f, 0.f, 0.f};

  for (int kt = 0; kt < 43; ++kt) {          // K = 676 -> 43 tiles (zero-pad)
    const int hwb = kt * 16;
#pragma unroll
    for (int e = 0; e < 2; ++e) {            // stage att 32x16
      int lin = e * 256 + tid;
      int mm = lin >> 4, kk = lin & 15;
      int hw = hwb + kk;
      S_a[mm][kk] = (hw < HW_) ? ab[mm * HW_ + hw] : 0.f;
    }
#pragma unroll
    for (int e = 0; e < 8; ++e) {            // stage + transpose F 128x16
      int lin = e * 256 + tid;
      int row = lin >> 4, col = lin & 15;    // row = local c, col = local hw
      int hw = hwb + col;
      S_f[col][row] = (hw < HW_) ? Fb[(size_t)row * HW_ + hw] : 0.f;
    }
    __syncthreads();
#pragma unroll
    for (int s = 0; s < 4; ++s) {
      const int kc = s * 4 + kh;
      v2f a;
      a.x = S_a[m0 + ln][kc];
      a.y = S_a[m0 + ln][kc + 1];
      v2f b0;
      b0.x = S_f[kc][ct0 * 16 + ln];
      b0.y = S_f[kc + 1][ct0 * 16 + ln];
      acc0 = wmma4(a, b0, acc0);
      v2f b1;
      b1.x = S_f[kc][(ct0 + 1) * 16 + ln];
      b1.y = S_f[kc + 1][(ct0 + 1) * 16 + ln];
      acc1 = wmma4(a, b1, acc1);
    }
    __syncthreads();
  }

  const float inv = 1.0f / (float)HW_;
  float ssq = 0.f;
  float* fo = fm_out + (size_t)b * KFC + cb * 128;
#pragma unroll
  for (int r = 0; r < 8; ++r) {
    const int m = m0 + r + ((lane >> 4) << 3);
    {
      float x = acc0[r] * inv;
      float sg = (x > 0.f) ? 1.f : ((x < 0.f) ? -1.f : 0.f);
      float sv = sg * sqrtf(fabsf(x) + EPSN);
      ssq += sv * sv;
      fo[(size_t)m * C_ + ct0 * 16 + ln] = sv;
    }
    {
      float x = acc1[r] * inv;
      float sg = (x > 0.f) ? 1.f : ((x < 0.f) ? -1.f : 0.f);
      float sv = sg * sqrtf(fabsf(x) + EPSN);
      ssq += sv * sv;
      fo[(size_t)m * C_ + (ct0 + 1) * 16 + ln] = sv;
    }
  }
  // deterministic reduction: wave shuffle -> LDS -> thread 0
#pragma unroll
  for (int off = 16; off > 0; off >>= 1) ssq += __shfl_xor(ssq, off, 32);
  if (lane == 0) red[w] = ssq;
  __syncthreads();
  if (tid == 0) {
    float t = 0.f;
#pragma unroll
    for (int i = 0; i < 8; ++i) t += red[i];
    ws[b * 6 + cb] = t;                      // exclusive slot per (b, cb)
  }
}

// =====================================================================
// Kernel 3: in-place global L2 normalize of fm per batch
// =====================================================================
__global__ __launch_bounds__(256) void k_norm(float* __restrict__ fm,
                                              const float* __restrict__ ws) {
  const int b = blockIdx.x;
  float s = 0.f;
#pragma unroll
  for (int i = 0; i < 6; ++i) s += ws[b * 6 + i];
  const float scale = 1.0f / fmaxf(sqrtf(s), EPSN);
  float* f = fm + (size_t)b * KFC;
  for (int i = threadIdx.x; i < KFC; i += 256) f[i] *= scale;
}

// =====================================================================
// Kernel 4: p (64x200) = 100 * fm_n (64x24576) @ fc_w^T + fc_b
// grid (13 class-tiles, 4 batch-tiles), block 256: 8 waves split K,
// per-wave LDS staging (fc_w transposed), deterministic LDS reduction.
// =====================================================================
__global__ __launch_bounds__(256) void k_fc(const float* __restrict__ fm,
                                            const float* __restrict__ Wc,
                                            const float* __restrict__ bc,
                                            float* __restrict__ p) {
  __shared__ float S_x[8][16][17];
  __shared__ float S_w[8][16][17];           // transposed: [k][n]
  __shared__ float red[8][256];
  const int tid  = threadIdx.x;
  const int lane = tid & 31;
  const int w    = tid >> 5;
  const int ln   = lane & 15;
  const int kh   = (lane >> 4) << 1;
  const int n0   = blockIdx.x * 16;          // class tile
  const int m0   = blockIdx.y * 16;          // batch tile
  const int ks   = w * (KFC / 8);            // 3072 per wave

  v8f acc = {0.f, 0.f, 0.f, 0.f, 0.f, 0.f, 0.f, 0.f};

  for (int kt = 0; kt < (KFC / 8) / 16; ++kt) {   // 192 k-tiles per wave
    const int kb = ks + kt * 16;
#pragma unroll
    for (int e = 0; e < 8; ++e) {            // per-wave private staging
      int lin = e * 32 + lane;
      int row = lin >> 4, col = lin & 15;
      S_x[w][row][col] = fm[(size_t)(m0 + row) * KFC + kb + col];
      float wv = (n0 + row < NCLS) ? Wc[(size_t)(n0 + row) * KFC + kb + col]
                                   : 0.f;
      S_w[w][col][row] = wv;                 // transpose into [k][n]
    }
    __syncthreads();
#pragma unroll
    for (int s = 0; s < 4; ++s) {
      const int kc = s * 4 + kh;
      v2f a;
      a.x = S_x[w][ln][kc];
      a.y = S_x[w][ln][kc + 1];
      v2f bv;
      bv.x = S_w[w][kc][ln];
      bv.y = S_w[w][kc + 1][ln];
      acc = wmma4(a, bv, acc);
    }
    __syncthreads();
  }

  // cross-wave K reduction, fixed order -> deterministic
#pragma unroll
  for (int r = 0; r < 8; ++r) red[r][w * 32 + lane] = acc[r];
  __syncthreads();
  if (w == 0) {
#pragma unroll
    for (int r = 0; r < 8; ++r) {
      float t = 0.f;
#pragma unroll
      for (int i = 0; i < 8; ++i) t += red[r][i * 32 + lane];
      const int m = m0 + r + ((lane >> 4) << 3);
      const int n = n0 + ln;
      if (n < NCLS) p[m * NCLS + n] = 100.f * t + bc[n];
    }
  }
}

// =====================================================================
extern "C" void kernel_launch(void* const* d_in, const int* in_sizes, int n_in,
                              void* d_out, int out_size, void* d_ws,
                              size_t ws_size, hipStream_t stream) {
  (void)in_sizes; (void)n_in; (void)out_size; (void)ws_size;
  const float* F    = (const float*)d_in[0];  // (64,768,26,26)
  const float* W    = (const float*)d_in[1];  // (192,768) -> rows 0..31 used
  const float* bias = (const float*)d_in[2];  // (192,)
  const float* fcw  = (const float*)d_in[3];  // (200,24576)
  const float* fcb  = (const float*)d_in[4];  // (200,)

  float* out = (float*)d_out;                 // [p | fm | att_m] flat
  float* p   = out;                           // 64*200
  float* fm  = out + (size_t)B_ * NCLS;       // 64*32*768
  float* att = fm + (size_t)B_ * M_ * C_;     // 64*32*676
  float* ws  = (float*)d_ws;                  // 64*6 partial sumsq

  k_att <<<dim3(64, 11), 256, 0, stream>>>(F, W, bias, att);
  k_fm  <<<dim3(64, 6),  256, 0, stream>>>(F, att, fm, ws);
  k_norm<<<64,           256, 0, stream>>>(fm, ws);
  k_fc  <<<dim3(13, 4),  256, 0, stream>>>(fm, fcw, fcb, p);
}